// BCEConnectivityLoss_82626580840507
// MI455X (gfx1250) — compile-verified
//
#include <hip/hip_runtime.h>

// ---------------------------------------------------------------------------
// BCEConnectivityLoss for MI455X (gfx1250, wave32)
//   out = mean(BCE(preds, targets)) + mean_b(components_b - 1)
// - All reductions via V_WMMA_F32_16X16X4_F32 (B = ones => rowsum engine),
//   cross-half combine via ds_swizzle SWAPX16. Deterministic (no f32 atomics).
// - CCL: reference-exact body; neighbor-min then double pointer jump, where
//   jump(jump(L))[i] == L[L[L[L[i]]]] on one materialized array.
// - All streaming kernels vectorized to b128 (4 elem/thread) for HBM.
// ---------------------------------------------------------------------------

typedef __attribute__((ext_vector_type(2))) float v2f;
typedef __attribute__((ext_vector_type(8))) float v8f;

#define IMG_H 512
#define IMG_W 512
#define IMG_N (IMG_H * IMG_W)        // 262144 = 2^18
#define IMG_MASK 0x3FFFF
#define CCL_BIG IMG_N                // background sentinel (per-image)
#define BATCH 64
#define TOTAL_N (BATCH * IMG_N)      // 16,777,216
#define CCL_ITERS 24
#define BCE_BLOCKS 4096
#define SWZ_SWAPX16 0x401f           // ds_swizzle: xor=0x10, or=0, and=0x1f

// ---- wave32 sum via WMMA f32 16x16x4 with B = ones --------------------------
__device__ __forceinline__ float wave_sum32(float s) {
#if defined(__AMDGCN__) && __has_builtin(__builtin_amdgcn_wmma_f32_16x16x4_f32)
  // A (16x4 f32): VGPR0 = {lanes0-15: K=0, lanes16-31: K=2}, VGPR1 = K=1/K=3.
  // Lane partial in a.x, zero in a.y; B = all ones.
  // D[m][n] = sum_k A[m][k] = s(lane m) + s(lane m+16).
  v2f a; a.x = s;    a.y = 0.0f;
  v2f b; b.x = 1.0f; b.y = 1.0f;
  v8f c = {};
  c = __builtin_amdgcn_wmma_f32_16x16x4_f32(
      /*neg_a=*/false, a, /*neg_b=*/false, b,
      /*c_mod=*/(short)0, c, /*reuse_a=*/false, /*reuse_b=*/false);
  // Lane l<16 holds D[0..7][l]; lane l+16 holds D[8..15][l].
  float t = c[0] + c[1] + c[2] + c[3] + c[4] + c[5] + c[6] + c[7];
  // Combine the two half-wave row groups: SWAPX16 swizzle (single DS op).
  t += __int_as_float(__builtin_amdgcn_ds_swizzle(__float_as_int(t),
                                                  SWZ_SWAPX16));
  return t;                     // full 32-lane sum, broadcast to all lanes
#else
  for (int m = 16; m >= 1; m >>= 1) s += __shfl_xor(s, m, 32);
  return s;
#endif
}

// Block (256 threads = 8 waves) sum; valid in thread 0.
__device__ __forceinline__ float block_sum256(float s, float* lds) {
  float w = wave_sum32(s);
  const int lane = threadIdx.x & 31;
  const int wid  = threadIdx.x >> 5;
  if (lane == 0) lds[wid] = w;
  __syncthreads();
  float r = 0.0f;
  if (threadIdx.x == 0) {
    #pragma unroll
    for (int i = 0; i < 8; ++i) r += lds[i];
  }
  __syncthreads();
  return r;
}

__device__ __forceinline__ int min5(int a, int b, int c, int d, int e) {
  return min(min(min(a, b), min(c, d)), e);
}

// ---- kernels ----------------------------------------------------------------

__global__ void zero_scalars_kernel(unsigned int* roots) {
  if (threadIdx.x == 0 && blockIdx.x == 0) *roots = 0u;
}

// BCE partial sums: one f32 per block, fixed-order (deterministic). b128 IO.
__global__ void bce_partial_kernel(const float* __restrict__ preds,
                                   const float* __restrict__ targets,
                                   float* __restrict__ partials, int nQuad) {
  __shared__ float lds[8];
  float s = 0.0f;
  const int stride = gridDim.x * blockDim.x;
  for (int q = blockIdx.x * blockDim.x + threadIdx.x; q < nQuad; q += stride) {
    const float4 p = *(const float4*)(preds + 4 * (size_t)q);
    const float4 t = *(const float4*)(targets + 4 * (size_t)q);
    #pragma unroll
    for (int k = 0; k < 4; ++k) {
      const float pk = (&p.x)[k];
      const float tk = (&t.x)[k];
      const float lp  = fmaxf(__logf(pk), -100.0f);          // torch clamp
      const float l1p = fmaxf(__logf(1.0f - pk), -100.0f);
      s = fmaf(-tk, lp, s);          // -= t*log(p)
      s = fmaf(tk - 1.0f, l1p, s);   // -= (1-t)*log(1-p)
    }
  }
  const float r = block_sum256(s, lds);
  if (threadIdx.x == 0) partials[blockIdx.x] = r;
}

// lab0 = idx if foreground else BIG (per-image local indices). 4 px/thread.
__global__ void init_labels_kernel(const float* __restrict__ preds,
                                   int* __restrict__ lab,
                                   int imgBase, int nQuad) {
  const int q = blockIdx.x * blockDim.x + threadIdx.x;
  if (q >= nQuad) return;
  const int g = 4 * q;
  const float4 p = *(const float4*)(preds + (size_t)imgBase * IMG_N + g);
  const int local = g & IMG_MASK;
  int4 o;
  o.x = (p.x != 0.0f) ? (local + 0) : CCL_BIG;
  o.y = (p.y != 0.0f) ? (local + 1) : CCL_BIG;
  o.z = (p.z != 0.0f) ? (local + 2) : CCL_BIG;
  o.w = (p.w != 0.0f) ? (local + 3) : CCL_BIG;
  *(int4*)(lab + g) = o;
}

// new = where(mask, min(lab, 4-neighbor mins), BIG). 4 px/thread; center and
// up/down rows come in as b128; horizontal neighbors mostly from registers.
__global__ void neighbor_min_kernel(const int* __restrict__ in,
                                    int* __restrict__ out, int nQuad) {
  const int q = blockIdx.x * blockDim.x + threadIdx.x;
  if (q >= nQuad) return;
  const int g = 4 * q;                       // 4 contiguous px, same row
  const int local = g & IMG_MASK;
  const int r  = local >> 9;                 // / W (W=512)
  const int c0 = local & (IMG_W - 1);        // 0..508, step 4

  const int4 v = *(const int4*)(in + g);
  int4 up = make_int4(CCL_BIG, CCL_BIG, CCL_BIG, CCL_BIG);
  int4 dn = up;
  if (r > 0)         up = *(const int4*)(in + g - IMG_W);
  if (r < IMG_H - 1) dn = *(const int4*)(in + g + IMG_W);
  const int lft = (c0 > 0)           ? in[g - 1] : CCL_BIG;
  const int rgt = (c0 < IMG_W - 4)   ? in[g + 4] : CCL_BIG;

  int4 o;
  o.x = (v.x >= CCL_BIG) ? CCL_BIG : min5(v.x, up.x, dn.x, lft, v.y);
  o.y = (v.y >= CCL_BIG) ? CCL_BIG : min5(v.y, up.y, dn.y, v.x, v.z);
  o.z = (v.z >= CCL_BIG) ? CCL_BIG : min5(v.z, up.z, dn.z, v.y, v.w);
  o.w = (v.w >= CCL_BIG) ? CCL_BIG : min5(v.w, up.w, dn.w, v.z, rgt);
  *(int4*)(out + g) = o;
}

// jump(jump(L))[i] == L[L[L[L[i]]]] : double pointer-jump as a 4-level chase.
// First level is a b128 load of 4 contiguous px; chases are scalar gathers.
__global__ void jump4_kernel(const int* __restrict__ in,
                             int* __restrict__ out, int nQuad) {
  const int q = blockIdx.x * blockDim.x + threadIdx.x;
  if (q >= nQuad) return;
  const int g = 4 * q;
  const int base = g & ~IMG_MASK;            // image base (IMG_N % 4 == 0)
  int4 v = *(const int4*)(in + g);
  #pragma unroll
  for (int k = 0; k < 4; ++k) {
    int x = (&v.x)[k];
    if (x < CCL_BIG) {                       // fg labels always < BIG
      x = in[base + x];
      x = in[base + x];
      x = in[base + x];
      (&v.x)[k] = x;
    } else {
      (&v.x)[k] = CCL_BIG;
    }
  }
  *(int4*)(out + g) = v;
}

// root <=> foreground && lab == own index; counts exact in f32 (< 2^24)
__global__ void count_roots_kernel(const int* __restrict__ lab,
                                   unsigned int* __restrict__ roots,
                                   int nQuad) {
  __shared__ float lds[8];
  float s = 0.0f;
  const int stride = gridDim.x * blockDim.x;
  for (int q = blockIdx.x * blockDim.x + threadIdx.x; q < nQuad; q += stride) {
    const int g = 4 * q;
    const int local = g & IMG_MASK;
    const int4 v = *(const int4*)(lab + g);
    s += (v.x == local + 0) ? 1.0f : 0.0f;   // roots are fg, so v < BIG
    s += (v.y == local + 1) ? 1.0f : 0.0f;
    s += (v.z == local + 2) ? 1.0f : 0.0f;
    s += (v.w == local + 3) ? 1.0f : 0.0f;
  }
  const float r = block_sum256(s, lds);
  if (threadIdx.x == 0) atomicAdd(roots, (unsigned int)(r + 0.5f));
}

// Fixed-order combine of per-block BCE partials + penalty -> out[0]
__global__ void finalize_kernel(const float* __restrict__ partials, int nPart,
                                const unsigned int* __restrict__ roots,
                                float* __restrict__ out) {
  __shared__ float lds[8];
  float s = 0.0f;
  for (int i = threadIdx.x; i < nPart; i += blockDim.x) s += partials[i];
  const float r = block_sum256(s, lds);
  if (threadIdx.x == 0) {
    const float bce = r / (float)TOTAL_N;
    const float penalty = (float)(*roots) / (float)BATCH - 1.0f;
    out[0] = bce + penalty;
  }
}

// ---------------------------------------------------------------------------

extern "C" void kernel_launch(void* const* d_in, const int* in_sizes, int n_in,
                              void* d_out, int out_size, void* d_ws, size_t ws_size,
                              hipStream_t stream) {
  (void)in_sizes; (void)n_in; (void)out_size;
  const float* preds   = (const float*)d_in[0];
  const float* targets = (const float*)d_in[1];
  float* out = (float*)d_out;

  // Workspace layout:
  //   [0, 4)        : u32 root count
  //   [256, 16640)  : 4096 f32 BCE block partials
  //   [16896, ...)  : ping/pong label buffers (chunked over batch if small)
  unsigned char* ws = (unsigned char*)d_ws;
  unsigned int* roots = (unsigned int*)ws;
  float* partials = (float*)(ws + 256);
  const size_t labOff = 16896;                 // 256-aligned
  const size_t perImg = (size_t)IMG_N * sizeof(int);

  int chunk = 1;
  if (ws_size > labOff + 2 * perImg) {
    size_t fit = (ws_size - labOff) / (2 * perImg);
    chunk = (fit > (size_t)BATCH) ? BATCH : (int)fit;
    if (chunk < 1) chunk = 1;
  }
  int* labA = (int*)(ws + labOff);
  int* labB = (int*)(ws + labOff + (size_t)chunk * perImg);

  zero_scalars_kernel<<<1, 32, 0, stream>>>(roots);

  bce_partial_kernel<<<BCE_BLOCKS, 256, 0, stream>>>(preds, targets, partials,
                                                     TOTAL_N / 4);

  for (int imgBase = 0; imgBase < BATCH; imgBase += chunk) {
    const int cn = (BATCH - imgBase < chunk) ? (BATCH - imgBase) : chunk;
    const int nQuad = cn * (IMG_N / 4);
    const int blocks = (nQuad + 255) / 256;

    init_labels_kernel<<<blocks, 256, 0, stream>>>(preds, labA, imgBase, nQuad);
    for (int it = 0; it < CCL_ITERS; ++it) {
      neighbor_min_kernel<<<blocks, 256, 0, stream>>>(labA, labB, nQuad);
      jump4_kernel<<<blocks, 256, 0, stream>>>(labB, labA, nQuad);
    }
    const int cblocks = blocks < 8192 ? blocks : 8192;
    count_roots_kernel<<<cblocks, 256, 0, stream>>>(labA, roots, nQuad);
  }

  finalize_kernel<<<1, 256, 0, stream>>>(partials, BCE_BLOCKS, roots, out);
}